// GNNAgent2_10514079941559
// MI455X (gfx1250) — compile-verified
//
#include <hip/hip_runtime.h>
#include <math.h>

typedef __attribute__((ext_vector_type(2))) float v2f;
typedef __attribute__((ext_vector_type(8))) float v8f;
typedef __attribute__((ext_vector_type(4))) int   v4i;

#define IN_DIM 96
#define H_DIM 128
#define A_DIM 30
#define ED_DIM 5
#define NEG_SLOPE 0.2f
#define KC 32              // K-chunk staged through LDS
#define LDSPAD 133         // row pitch (floats): 133 mod 64 coprime -> no bank conflicts

// ---- gfx1250 async global->LDS path (ASYNCcnt), guarded so compile never breaks
#if defined(__has_builtin)
#if __has_builtin(__builtin_amdgcn_global_load_async_to_lds_b128)
#define HAVE_ASYNC_LDS 1
#endif
#endif

// builtin takes int4-vector pointers (per hipcc diagnostic): global src, LDS dst
typedef __attribute__((address_space(1))) v4i* gptr_v4i;
typedef __attribute__((address_space(3))) v4i* lptr_v4i;

__device__ __forceinline__ void wait_async_lds() {
#if defined(HAVE_ASYNC_LDS)
#if __has_builtin(__builtin_amdgcn_s_wait_asynccnt)
  __builtin_amdgcn_s_wait_asynccnt(0);
#else
  asm volatile("s_wait_asynccnt 0" ::: "memory");
#endif
#endif
}

// ---------------------------------------------------------------------------
// float atomic max via signed/unsigned integer atomics (location init to -inf)
// ---------------------------------------------------------------------------
__device__ __forceinline__ void atomicMaxF(float* addr, float v) {
  if (v >= 0.0f) atomicMax((int*)addr, __float_as_int(v));
  else           atomicMin((unsigned int*)addr, __float_as_uint(v));
}

// ---------------------------------------------------------------------------
// WMMA fp32 GEMM:  C[M x Nc] = act(A[M x K] @ B[K x Nc] + bias),  Nc <= 128
// One wave -> one 16x16 C tile via V_WMMA_F32_16X16X4_F32.
// blockDim.x/32 waves cover consecutive 16-col tiles; gridDim.x = M/16 tiles.
// A tile (16 x K) in LDS once; B staged in KC-row chunks (async-to-LDS when
// the panel is full width). Inner loop is pure ds_load + v_wmma, no branches.
// ISA VGPR layouts (CDNA5 ISA §7.12.2):
//   A 16x4 f32: lanes0-15 rows M; VGPR0 = K{0|2}, VGPR1 = K{1|3} (half-split)
//   C/D 16x16 f32: VGPR r -> M = r (+8 for lanes 16-31), N = lane&15
// ---------------------------------------------------------------------------
__global__ void wmma_gemm_f32_kernel(const float* __restrict__ A, int lda,
                                     const float* __restrict__ B, int ldb,
                                     const float* __restrict__ bias,
                                     float* __restrict__ C, int ldc,
                                     int M, int Nc, int K, int relu)
{
  __shared__ float lA[16][LDSPAD];
  __shared__ float sB[KC][LDSPAD];

  const int tid  = threadIdx.x;
  const int wave = tid >> 5;
  const int lane = tid & 31;
  const int half = lane >> 4;
  const int l16  = lane & 15;
  const int m0   = blockIdx.x * 16;
  const int n0   = wave * 16;

  // ---- stage the 16 x K A tile (K multiple of 4, rows 16B aligned) ----
  const int nv = K >> 2;
  for (int i = tid; i < 16 * nv; i += blockDim.x) {
    const int r = i / nv, c = i % nv;
    const float4 v = reinterpret_cast<const float4*>(A + (size_t)(m0 + r) * lda)[c];
    lA[r][4 * c + 0] = v.x; lA[r][4 * c + 1] = v.y;
    lA[r][4 * c + 2] = v.z; lA[r][4 * c + 3] = v.w;
  }

  const int col = n0 + l16;
  v8f acc = {};

  for (int kc = 0; kc < K; kc += KC) {
    __syncthreads();   // A tile ready (1st iter) / previous chunk consumed

    if (Nc == H_DIM) {
      // full 32 x 128 panel: one b128 per thread-iteration
      for (int i = tid; i < KC * (H_DIM / 4); i += blockDim.x) {
        const int r = i >> 5, c4 = (i & 31) << 2;
        const float* g = B + (size_t)(kc + r) * ldb + c4;
#if defined(HAVE_ASYNC_LDS)
        __builtin_amdgcn_global_load_async_to_lds_b128(
            (gptr_v4i)(void*)g, (lptr_v4i)(void*)&sB[r][c4], 0, 0);
#else
        const float4 v = *reinterpret_cast<const float4*>(g);
        sB[r][c4 + 0] = v.x; sB[r][c4 + 1] = v.y;
        sB[r][c4 + 2] = v.z; sB[r][c4 + 3] = v.w;
#endif
      }
      wait_async_lds();
    } else {
      // narrow panel (e.g. Nc = 30): pad columns to 32 with zeros so the
      // K-loop needs no column predicate
      for (int i = tid; i < KC * 32; i += blockDim.x) {
        const int r = i >> 5, c = i & 31;
        sB[r][c] = (c < Nc) ? B[(size_t)(kc + r) * ldb + c] : 0.0f;
      }
    }
    __syncthreads();   // B chunk visible to all waves

#pragma unroll
    for (int kk = 0; kk < KC; kk += 4) {
      v2f a, b;
      a.x = lA[l16][kc + kk + 2 * half + 0];
      a.y = lA[l16][kc + kk + 2 * half + 1];
      b.x = sB[kk + 2 * half + 0][col];
      b.y = sB[kk + 2 * half + 1][col];
      acc = __builtin_amdgcn_wmma_f32_16x16x4_f32(false, a, false, b,
                                                  (short)0, acc, false, false);
    }
  }

  if (col < Nc) {
    const float bs = bias ? bias[col] : 0.0f;
#pragma unroll
    for (int r = 0; r < 8; ++r) {
      const int row = m0 + r + 8 * half;
      if (row < M) {
        float v = acc[r] + bs;
        if (relu) v = fmaxf(v, 0.0f);
        C[(size_t)row * ldc + col] = v;
      }
    }
  }
}

// ---------------------------------------------------------------------------
// per-node init: deg = 0, attr_sum = 0, smax = -inf, denom = 0
// ---------------------------------------------------------------------------
__global__ void init_node_kernel(float* __restrict__ deg, float* __restrict__ attrs,
                                 float* __restrict__ smax, float* __restrict__ denom,
                                 int n)
{
  const int i = blockIdx.x * blockDim.x + threadIdx.x;
  if (i >= n) return;
  deg[i]   = 0.0f;
  smax[i]  = -INFINITY;
  denom[i] = 0.0f;
#pragma unroll
  for (int j = 0; j < ED_DIM; ++j) attrs[i * ED_DIM + j] = 0.0f;
}

// ---------------------------------------------------------------------------
// in-degree + incoming edge-attr sums (for self-loop fill_value='mean')
// ---------------------------------------------------------------------------
__global__ void edge_deg_attr_kernel(const int* __restrict__ dst,
                                     const float* __restrict__ eattr,
                                     float* __restrict__ deg,
                                     float* __restrict__ attrs, int E)
{
  const int e = blockIdx.x * blockDim.x + threadIdx.x;
  if (e >= E) return;
  const int d = dst[e];
  atomicAdd(&deg[d], 1.0f);
#pragma unroll
  for (int j = 0; j < ED_DIM; ++j)
    atomicAdd(&attrs[(size_t)d * ED_DIM + j], eattr[(size_t)e * ED_DIM + j]);
}

__global__ void loop_attr_kernel(float* __restrict__ attrs,
                                 const float* __restrict__ deg, int n)
{
  const int i = blockIdx.x * blockDim.x + threadIdx.x;
  if (i >= n) return;
  const float inv = 1.0f / fmaxf(deg[i], 1.0f);
#pragma unroll
  for (int j = 0; j < ED_DIM; ++j) attrs[(size_t)i * ED_DIM + j] *= inv;
}

// ---------------------------------------------------------------------------
// wave-per-edge attention score + segment max.
// edges [0,E): from edge_index; edges [E,total): self loops with mean attrs.
// ---------------------------------------------------------------------------
__global__ void edge_score_kernel(const int* __restrict__ src, const int* __restrict__ dst,
                                  const float* __restrict__ eattr,    // E x ED
                                  const float* __restrict__ lattr,    // N x ED (self-loop)
                                  const float* __restrict__ x_l, const float* __restrict__ x_r,
                                  const float* __restrict__ W_edge,   // ED x H
                                  const float* __restrict__ att,      // H
                                  float* __restrict__ score, float* __restrict__ smax,
                                  int E, int total)
{
  __shared__ float sWe[ED_DIM * H_DIM];
  __shared__ float sAtt[H_DIM];
  for (int i = threadIdx.x; i < ED_DIM * H_DIM; i += blockDim.x) sWe[i] = W_edge[i];
  for (int i = threadIdx.x; i < H_DIM; i += blockDim.x) sAtt[i] = att[i];
  __syncthreads();

  const int lane = threadIdx.x & 31;
  const int e = blockIdx.x * (blockDim.x >> 5) + (threadIdx.x >> 5);
  if (e >= total) return;

  int s, d; const float* ea;
  if (e < E) { s = src[e]; d = dst[e]; ea = eattr + (size_t)e * ED_DIM; }
  else       { s = e - E;  d = s;      ea = lattr + (size_t)s * ED_DIM; }
  const float e0 = ea[0], e1 = ea[1], e2 = ea[2], e3 = ea[3], e4 = ea[4];

  const int c = lane * 4;
  const float4 xl = *reinterpret_cast<const float4*>(x_l + (size_t)s * H_DIM + c);
  const float4 xr = *reinterpret_cast<const float4*>(x_r + (size_t)d * H_DIM + c);
  const float lf[4] = {xl.x, xl.y, xl.z, xl.w};
  const float rf[4] = {xr.x, xr.y, xr.z, xr.w};

  float part = 0.0f;
#pragma unroll
  for (int j = 0; j < 4; ++j) {
    const int cc = c + j;
    float m = lf[j] + rf[j]
            + e0 * sWe[0 * H_DIM + cc] + e1 * sWe[1 * H_DIM + cc]
            + e2 * sWe[2 * H_DIM + cc] + e3 * sWe[3 * H_DIM + cc]
            + e4 * sWe[4 * H_DIM + cc];
    m = (m > 0.0f) ? m : NEG_SLOPE * m;      // leaky_relu
    part += m * sAtt[cc];
  }
#pragma unroll
  for (int off = 16; off > 0; off >>= 1) part += __shfl_xor(part, off, 32);

  if (lane == 0) { score[e] = part; atomicMaxF(&smax[d], part); }
}

// ---------------------------------------------------------------------------
// softmax denominator per destination node
// ---------------------------------------------------------------------------
__global__ void edge_denom_kernel(const int* __restrict__ dst,
                                  const float* __restrict__ score,
                                  const float* __restrict__ smax,
                                  float* __restrict__ denom, int E, int total)
{
  const int e = blockIdx.x * blockDim.x + threadIdx.x;
  if (e >= total) return;
  const int d = (e < E) ? dst[e] : e - E;
  atomicAdd(&denom[d], __expf(score[e] - smax[d]));
}

// ---------------------------------------------------------------------------
// wave-per-edge weighted scatter: h[dst] += alpha * x_l[src]
// (h pre-initialized with x @ res_W + gat_bias by the residual GEMM)
// ---------------------------------------------------------------------------
__global__ void edge_accum_kernel(const int* __restrict__ src, const int* __restrict__ dst,
                                  const float* __restrict__ score,
                                  const float* __restrict__ smax,
                                  const float* __restrict__ denom,
                                  const float* __restrict__ x_l,
                                  float* __restrict__ h, int E, int total)
{
  const int lane = threadIdx.x & 31;
  const int e = blockIdx.x * (blockDim.x >> 5) + (threadIdx.x >> 5);
  if (e >= total) return;

  int s, d;
  if (e < E) { s = src[e]; d = dst[e]; }
  else       { s = e - E;  d = s; }
  const float coef = __expf(score[e] - smax[d]) / denom[d];

  const int c = lane * 4;
  const float4 xl = *reinterpret_cast<const float4*>(x_l + (size_t)s * H_DIM + c);
  float* hp = h + (size_t)d * H_DIM + c;
  atomicAdd(hp + 0, coef * xl.x);
  atomicAdd(hp + 1, coef * xl.y);
  atomicAdd(hp + 2, coef * xl.z);
  atomicAdd(hp + 3, coef * xl.w);
}

// ---------------------------------------------------------------------------
// launch
// ---------------------------------------------------------------------------
extern "C" void kernel_launch(void* const* d_in, const int* in_sizes, int n_in,
                              void* d_out, int out_size, void* d_ws, size_t ws_size,
                              hipStream_t stream)
{
  const float* inputs   = (const float*)d_in[0];
  // d_in[1] hidden_states + d_in[6..9] GRU params unused: reference discards _h_gru
  const int*   eidx     = (const int*)d_in[2];
  const float* eattr    = (const float*)d_in[3];
  const float* W_e1     = (const float*)d_in[4];
  const float* b_e1     = (const float*)d_in[5];
  const float* W_l      = (const float*)d_in[10];
  const float* b_l      = (const float*)d_in[11];
  const float* W_r      = (const float*)d_in[12];
  const float* b_r      = (const float*)d_in[13];
  const float* W_edge   = (const float*)d_in[14];
  const float* att      = (const float*)d_in[15];
  const float* res_W    = (const float*)d_in[16];
  const float* gat_bias = (const float*)d_in[17];
  const float* W_fc2    = (const float*)d_in[18];
  const float* b_fc2    = (const float*)d_in[19];

  const int N     = in_sizes[0] / IN_DIM;   // 50000
  const int E     = in_sizes[2] / 2;        // 600000
  const int total = E + N;
  const int* src  = eidx;
  const int* dst  = eidx + E;

  float* q = (float*)d_out;                 // N x A
  float* h = q + (size_t)N * A_DIM;         // N x H (accumulated in place)

  float* ws    = (float*)d_ws;
  float* x     = ws;  ws += (size_t)N * H_DIM;
  float* x_l   = ws;  ws += (size_t)N * H_DIM;
  float* x_r   = ws;  ws += (size_t)N * H_DIM;
  float* deg   = ws;  ws += N;
  float* attrs = ws;  ws += (size_t)N * ED_DIM;
  float* smax  = ws;  ws += N;
  float* denom = ws;  ws += N;
  float* score = ws;  ws += total;

  const int mtiles = (N + 15) / 16;

  // per-node accumulator init
  init_node_kernel<<<(N + 255) / 256, 256, 0, stream>>>(deg, attrs, smax, denom, N);

  // x = relu(inputs @ W_e1 + b_e1)
  wmma_gemm_f32_kernel<<<mtiles, 256, 0, stream>>>(inputs, IN_DIM, W_e1, H_DIM, b_e1,
                                                   x, H_DIM, N, H_DIM, IN_DIM, 1);
  // x_l = x @ W_l + b_l ; x_r = x @ W_r + b_r ; h = x @ res_W + gat_bias
  wmma_gemm_f32_kernel<<<mtiles, 256, 0, stream>>>(x, H_DIM, W_l, H_DIM, b_l,
                                                   x_l, H_DIM, N, H_DIM, H_DIM, 0);
  wmma_gemm_f32_kernel<<<mtiles, 256, 0, stream>>>(x, H_DIM, W_r, H_DIM, b_r,
                                                   x_r, H_DIM, N, H_DIM, H_DIM, 0);
  wmma_gemm_f32_kernel<<<mtiles, 256, 0, stream>>>(x, H_DIM, res_W, H_DIM, gat_bias,
                                                   h, H_DIM, N, H_DIM, H_DIM, 0);

  // self-loop mean edge attrs
  edge_deg_attr_kernel<<<(E + 255) / 256, 256, 0, stream>>>(dst, eattr, deg, attrs, E);
  loop_attr_kernel<<<(N + 255) / 256, 256, 0, stream>>>(attrs, deg, N);

  // attention: score + segment max -> denom -> weighted scatter into h
  const int wpb  = 8;                               // waves per block
  const int blkE = (total + wpb - 1) / wpb;
  edge_score_kernel<<<blkE, 32 * wpb, 0, stream>>>(src, dst, eattr, attrs, x_l, x_r,
                                                   W_edge, att, score, smax, E, total);
  edge_denom_kernel<<<(total + 255) / 256, 256, 0, stream>>>(dst, score, smax, denom, E, total);
  edge_accum_kernel<<<blkE, 32 * wpb, 0, stream>>>(src, dst, score, smax, denom, x_l, h, E, total);

  // q = h @ W_fc2 + b_fc2  (2 waves cover the 30 columns)
  wmma_gemm_f32_kernel<<<mtiles, 64, 0, stream>>>(h, H_DIM, W_fc2, A_DIM, b_fc2,
                                                  q, A_DIM, N, A_DIM, H_DIM, 0);
}